// VariationalGCNEncoder_62904091018060
// MI455X (gfx1250) — compile-verified
//
#include <hip/hip_runtime.h>
#include <hip/hip_bf16.h>

typedef float v2f __attribute__((ext_vector_type(2)));
typedef float v8f __attribute__((ext_vector_type(8)));

// ---------------------------------------------------------------------------
// Degree kernels
// ---------------------------------------------------------------------------
__global__ __launch_bounds__(256) void deg_init(float* __restrict__ deg, int N) {
    int i = blockIdx.x * blockDim.x + threadIdx.x;
    if (i < N) deg[i] = 1.0f;  // self-loop contribution
}

__global__ __launch_bounds__(256) void deg_count(const int* __restrict__ dst,
                                                 float* __restrict__ deg, int E) {
    int i = blockIdx.x * blockDim.x + threadIdx.x;
    if (i < E) atomicAdd(&deg[dst[i]], 1.0f);
}

__global__ __launch_bounds__(256) void deg_rsqrt(float* __restrict__ deg, int N) {
    int i = blockIdx.x * blockDim.x + threadIdx.x;
    if (i < N) deg[i] = rsqrtf(deg[i]);
}

// ---------------------------------------------------------------------------
// FP32 WMMA GEMM: C[M,Nc] = A[M,K] @ B[K,Nc], row-major.
// One wave -> one 64x16 macro-tile (4 stacked 16x16 C tiles sharing the B
// fragment), K consumed 4 at a time via V_WMMA_F32_16X16X4_F32.
// Requires K % 4 == 0, Nc % 16 == 0.
// Fragment layouts per CDNA5 ISA 7.12.2:
//   A 16x4:  lane l holds row (l&15); VGPR0/1 = K = koff, koff+1; koff=(l>=16)?2:0
//   B 4x16:  VGPR v holds K-row (v + koff) at column n0 + (l&15)
//   C 16x16: VGPR v holds row m0 + v + ((l>=16)?8:0) at column n0 + (l&15)
// Four independent accumulators per k-step -> no WMMA->WMMA RAW chain.
// ---------------------------------------------------------------------------
__global__ __launch_bounds__(256) void gemm_wmma_f32(const float* __restrict__ A,
                                                     const float* __restrict__ B,
                                                     float* __restrict__ C,
                                                     int M, int Nc, int K) {
    const int lane = threadIdx.x & 31;
    const int wave = threadIdx.x >> 5;
    const int tile = blockIdx.x * (blockDim.x >> 5) + wave;
    const int tilesN = Nc >> 4;
    const int tilesM = (M + 63) >> 6;          // 64-row macro tiles
    if (tile >= tilesM * tilesN) return;       // wave-uniform exit

    const int tm = tile / tilesN;
    const int tn = tile - tm * tilesN;
    const int m0 = tm << 6;
    const int n0 = tn << 4;

    const int lrow = lane & 15;
    const int koff = (lane >> 4) << 1;         // 0 or 2
    const int col  = n0 + lrow;
    const int Mm1  = M - 1;

    // Clamped row pointers for the 4 stacked tiles (tail-safe, no OOB).
    const float* __restrict__ ap0 = A + (size_t)min(m0 +  0 + lrow, Mm1) * K;
    const float* __restrict__ ap1 = A + (size_t)min(m0 + 16 + lrow, Mm1) * K;
    const float* __restrict__ ap2 = A + (size_t)min(m0 + 32 + lrow, Mm1) * K;
    const float* __restrict__ ap3 = A + (size_t)min(m0 + 48 + lrow, Mm1) * K;

    v8f acc0 = {}, acc1 = {}, acc2 = {}, acc3 = {};
    for (int k0 = 0; k0 < K; k0 += 4) {
        // B fragment: two K-rows at this column (shared by all 4 tiles)
        const float* bp = B + (size_t)(k0 + koff) * Nc + col;
        v2f b;
        b.x = bp[0];
        b.y = bp[Nc];
        // A fragments: contiguous float2 (8B aligned: k0+koff even)
        v2f a0 = *(const v2f*)(ap0 + k0 + koff);
        v2f a1 = *(const v2f*)(ap1 + k0 + koff);
        v2f a2 = *(const v2f*)(ap2 + k0 + koff);
        v2f a3 = *(const v2f*)(ap3 + k0 + koff);
        acc0 = __builtin_amdgcn_wmma_f32_16x16x4_f32(false, a0, false, b,
                                                     (short)0, acc0, false, false);
        acc1 = __builtin_amdgcn_wmma_f32_16x16x4_f32(false, a1, false, b,
                                                     (short)0, acc1, false, false);
        acc2 = __builtin_amdgcn_wmma_f32_16x16x4_f32(false, a2, false, b,
                                                     (short)0, acc2, false, false);
        acc3 = __builtin_amdgcn_wmma_f32_16x16x4_f32(false, a3, false, b,
                                                     (short)0, acc3, false, false);
    }

    // Store the 4 stacked 16x16 tiles (guarded for the ragged last macro-tile).
    const int hi = (lane >> 4) << 3;           // +0 or +8
    v8f accs[4] = {acc0, acc1, acc2, acc3};
#pragma unroll
    for (int t = 0; t < 4; ++t) {
        const int rbase = m0 + (t << 4) + hi;
#pragma unroll
        for (int v = 0; v < 8; ++v) {
            int r = rbase + v;
            if (r < M) C[(size_t)r * Nc + col] = accs[t][v];
        }
    }
}

// ---------------------------------------------------------------------------
// Edge scatter: one wave per edge (E real edges, then N self-loops).
// out[dst] += (dis[src]*dis[dst]) * t[src].  F multiple of 4, F <= 128.
// Lane handles 4 consecutive floats (float4 gather + 4 f32 atomics).
// ---------------------------------------------------------------------------
__global__ __launch_bounds__(256) void gcn_scatter(const float* __restrict__ t,
                                                   const int* __restrict__ src,
                                                   const int* __restrict__ dst,
                                                   const float* __restrict__ dis,
                                                   float* __restrict__ out,
                                                   int E, int N, int F) {
    int w = (blockIdx.x * blockDim.x + threadIdx.x) >> 5;
    int lane = threadIdx.x & 31;
    if (w >= E + N) return;
    int s, d;
    if (w < E) { s = src[w]; d = dst[w]; }
    else       { s = d = w - E; }
    float norm = dis[s] * dis[d];
    const float* tp = t + (size_t)s * F;
    float* op = out + (size_t)d * F;
    for (int f = lane * 4; f < F; f += 128) {
        float4 v = *(const float4*)(tp + f);
        atomicAdd(op + f + 0, v.x * norm);
        atomicAdd(op + f + 1, v.y * norm);
        atomicAdd(op + f + 2, v.z * norm);
        atomicAdd(op + f + 3, v.w * norm);
    }
}

// Dual scatter for layer 2: lanes 0-15 handle mu (t2), lanes 16-31 logvar (t3).
// F = F_OUT (64): 16 lanes x 4 floats covers each output.
__global__ __launch_bounds__(256) void gcn_scatter2(const float* __restrict__ t2,
                                                    const float* __restrict__ t3,
                                                    const int* __restrict__ src,
                                                    const int* __restrict__ dst,
                                                    const float* __restrict__ dis,
                                                    float* __restrict__ mu,
                                                    float* __restrict__ lv,
                                                    int E, int N, int F) {
    int w = (blockIdx.x * blockDim.x + threadIdx.x) >> 5;
    int lane = threadIdx.x & 31;
    if (w >= E + N) return;
    int s, d;
    if (w < E) { s = src[w]; d = dst[w]; }
    else       { s = d = w - E; }
    float norm = dis[s] * dis[d];
    int half = lane >> 4;
    int l    = lane & 15;
    const float* tp = (half ? t3 : t2) + (size_t)s * F;
    float* op       = (half ? lv : mu) + (size_t)d * F;
    for (int f = l * 4; f < F; f += 64) {
        float4 v = *(const float4*)(tp + f);
        atomicAdd(op + f + 0, v.x * norm);
        atomicAdd(op + f + 1, v.y * norm);
        atomicAdd(op + f + 2, v.z * norm);
        atomicAdd(op + f + 3, v.w * norm);
    }
}

// ---------------------------------------------------------------------------
// Bias + LeakyReLU (layer 1), bias-only (layer 2)
// ---------------------------------------------------------------------------
__global__ __launch_bounds__(256) void bias_lrelu(float* __restrict__ h,
                                                  const float* __restrict__ b,
                                                  long long total, int F) {
    long long i = (long long)blockIdx.x * blockDim.x + threadIdx.x;
    if (i >= total) return;
    float v = h[i] + b[(int)(i % F)];
    h[i] = (v > 0.0f) ? v : 0.01f * v;
}

__global__ __launch_bounds__(256) void bias_add2(float* __restrict__ mu,
                                                 float* __restrict__ lv,
                                                 const float* __restrict__ bmu,
                                                 const float* __restrict__ blv,
                                                 long long total, int F) {
    long long i = (long long)blockIdx.x * blockDim.x + threadIdx.x;
    if (i >= total) return;
    int f = (int)(i % F);
    mu[i] += bmu[f];
    lv[i] += blv[f];
}

// ---------------------------------------------------------------------------
// Host launcher
// ---------------------------------------------------------------------------
extern "C" void kernel_launch(void* const* d_in, const int* in_sizes, int n_in,
                              void* d_out, int out_size, void* d_ws, size_t ws_size,
                              hipStream_t stream) {
    const float* x  = (const float*)d_in[0];
    const int*   ei = (const int*)d_in[1];
    const float* W1 = (const float*)d_in[2];
    const float* b1 = (const float*)d_in[3];
    const float* Wm = (const float*)d_in[4];
    const float* bm = (const float*)d_in[5];
    const float* Wl = (const float*)d_in[6];
    const float* bl = (const float*)d_in[7];

    const int F_HID = in_sizes[3];              // 128
    const int F_IN  = in_sizes[2] / F_HID;      // 256
    const int N     = in_sizes[0] / F_IN;       // 50000
    const int F_OUT = in_sizes[5];              // 64
    const int E     = in_sizes[1] / 2;          // 800000
    const int* src = ei;
    const int* dst = ei + E;

    // workspace layout (floats): dis[N] | buf1[N*F_HID] | h1[N*F_HID]
    float* ws   = (float*)d_ws;
    float* dis  = ws;
    float* buf1 = ws + N;
    float* h1   = buf1 + (size_t)N * F_HID;
    float* t1 = buf1;                       // layer-1 pre-aggregation
    float* t2 = buf1;                       // reuse after agg1 (2*F_OUT <= F_HID)
    float* t3 = buf1 + (size_t)N * F_OUT;
    float* mu = (float*)d_out;
    float* lv = mu + (size_t)N * F_OUT;

    // degrees -> deg^{-1/2}
    deg_init <<<(N + 255) / 256, 256, 0, stream>>>(dis, N);
    deg_count<<<(E + 255) / 256, 256, 0, stream>>>(dst, dis, E);
    deg_rsqrt<<<(N + 255) / 256, 256, 0, stream>>>(dis, N);

    const long long waves  = (long long)E + N;
    const int scatterBlks  = (int)((waves + 7) / 8);

    // layer 1: h1 = lrelu(scatter(norm * (x@W1)) + b1)
    {
        int tiles = ((N + 63) / 64) * (F_HID / 16);
        gemm_wmma_f32<<<(tiles + 7) / 8, 256, 0, stream>>>(x, W1, t1, N, F_HID, F_IN);
    }
    hipMemsetAsync(h1, 0, (size_t)N * F_HID * sizeof(float), stream);
    gcn_scatter<<<scatterBlks, 256, 0, stream>>>(t1, src, dst, dis, h1, E, N, F_HID);
    {
        long long total = (long long)N * F_HID;
        bias_lrelu<<<(int)((total + 255) / 256), 256, 0, stream>>>(h1, b1, total, F_HID);
    }

    // layer 2: mu / logvar
    {
        int tiles = ((N + 63) / 64) * (F_OUT / 16);
        gemm_wmma_f32<<<(tiles + 7) / 8, 256, 0, stream>>>(h1, Wm, t2, N, F_OUT, F_HID);
        gemm_wmma_f32<<<(tiles + 7) / 8, 256, 0, stream>>>(h1, Wl, t3, N, F_OUT, F_HID);
    }
    hipMemsetAsync(d_out, 0, (size_t)2 * N * F_OUT * sizeof(float), stream);
    gcn_scatter2<<<scatterBlks, 256, 0, stream>>>(t2, t3, src, dst, dis, mu, lv, E, N, F_OUT);
    {
        long long total = (long long)N * F_OUT;
        bias_add2<<<(int)((total + 255) / 256), 256, 0, stream>>>(mu, lv, bm, bl, total, F_OUT);
    }
}